// Net_49203145343456
// MI455X (gfx1250) — compile-verified
//
#include <hip/hip_runtime.h>
#include <math.h>

// ---------------------------------------------------------------------------
// GraphSAGE 2-layer forward for MI455X (gfx1250, wave32).
// GEMMs use V_WMMA_F32_16X16X4_F32 (full fp32 — matmuls are a small fraction
// of total time vs. the HBM/atomic-bound edge aggregation, so low precision
// buys nothing here).
// ---------------------------------------------------------------------------

#define N_NODES 50000
#define N_EDGES 800000
#define IN_CH   128
#define HIDDEN  128
#define OUT_CH  40

typedef __attribute__((ext_vector_type(2))) float v2f;
typedef __attribute__((ext_vector_type(8))) float v8f;

// ---------------- zero fill (float4 stores) ----------------
__global__ __launch_bounds__(256) void zero_kernel(float4* __restrict__ p, int n4) {
    int i = blockIdx.x * 256 + threadIdx.x;
    if (i < n4) p[i] = make_float4(0.f, 0.f, 0.f, 0.f);
}

// ---------------- edge scatter-add: one wave per edge ----------------
// agg[dst] += feat[src]; optionally cnt[dst] += 1 (lane 0).
__global__ __launch_bounds__(256) void scatter_kernel(
    const int* __restrict__ srcIdx, const int* __restrict__ dstIdx,
    const float* __restrict__ feat, float* __restrict__ agg,
    float* __restrict__ cnt, int nEdges)
{
    int tid  = blockIdx.x * 256 + threadIdx.x;
    int e    = tid >> 5;
    if (e >= nEdges) return;
    int lane = tid & 31;
    int s = srcIdx[e];
    int d = dstIdx[e];
    const float* f = feat + (long)s * 128;
    float*       a = agg  + (long)d * 128;
    if (cnt != nullptr && lane == 0) atomicAdd(cnt + d, 1.0f);
#pragma unroll
    for (int i = 0; i < 4; ++i) {
        int c = lane + 32 * i;
        atomicAdd(a + c, f[c]);   // global_atomic_add_f32, L2-resident
    }
}

// ---------------- layer 1 GEMM: h = relu([mean | x] @ [Wl;Wr] + b) ----------
// One wave per 16-row M-tile; 8 N-tiles of 16 (HIDDEN=128); K=256 in steps of 4.
__global__ __launch_bounds__(256) void gemm1_kernel(
    const float* __restrict__ agg, const float* __restrict__ cnt,
    const float* __restrict__ x,
    const float* __restrict__ Wl, const float* __restrict__ bl,
    const float* __restrict__ Wr, float* __restrict__ h)
{
    int wave = (blockIdx.x * 256 + threadIdx.x) >> 5;
    if (wave >= N_NODES / 16) return;          // wave-uniform: EXEC stays full
    int lane = threadIdx.x & 31;
    int half = lane >> 4;                      // 0: K+0..1, 1: K+2..3 (A frag)
    int l16  = lane & 15;

    int row = wave * 16 + l16;                 // A-matrix row for this lane
    float scale = 1.0f / fmaxf(cnt[row], 1.0f);
    const float2* arow = (const float2*)(agg + (long)row * 128);
    const float2* xrow = (const float2*)(x   + (long)row * 128);

    v8f c[8];
#pragma unroll
    for (int nt = 0; nt < 8; ++nt) {
        float bv = bl[nt * 16 + l16];          // bias broadcast down rows
#pragma unroll
        for (int r = 0; r < 8; ++r) c[nt][r] = bv;
    }

    // Phase 1: mean part (agg/cnt) against Wl  (K = 0..127)
    for (int k0 = 0; k0 < 128; k0 += 4) {
        int ka = k0 + 2 * half;
        float2 av = arow[ka >> 1];
        v2f a; a.x = av.x * scale; a.y = av.y * scale;
#pragma unroll
        for (int nt = 0; nt < 8; ++nt) {
            v2f b;
            b.x = Wl[ka * 128 + nt * 16 + l16];
            b.y = Wl[(ka + 1) * 128 + nt * 16 + l16];
            c[nt] = __builtin_amdgcn_wmma_f32_16x16x4_f32(
                false, a, false, b, (short)0, c[nt], false, false);
        }
    }
    // Phase 2: self part (x) against Wr  (K = 128..255)
    for (int k0 = 0; k0 < 128; k0 += 4) {
        int ka = k0 + 2 * half;
        float2 av = xrow[ka >> 1];
        v2f a; a.x = av.x; a.y = av.y;
#pragma unroll
        for (int nt = 0; nt < 8; ++nt) {
            v2f b;
            b.x = Wr[ka * 128 + nt * 16 + l16];
            b.y = Wr[(ka + 1) * 128 + nt * 16 + l16];
            c[nt] = __builtin_amdgcn_wmma_f32_16x16x4_f32(
                false, a, false, b, (short)0, c[nt], false, false);
        }
    }

    // Store with fused ReLU. C VGPR r: rows (r + 8*half), col = nt*16 + l16.
    int rbase = wave * 16 + 8 * half;
#pragma unroll
    for (int nt = 0; nt < 8; ++nt)
#pragma unroll
        for (int r = 0; r < 8; ++r)
            h[(long)(rbase + r) * 128 + nt * 16 + l16] = fmaxf(c[nt][r], 0.0f);
}

// ---------------- layer 2 GEMM: logits = [mean2 | h] @ [Wl2;Wr2] + b2 -------
// N = 40 -> 3 N-tiles of 16 (last padded with zeros), K = 256.
__global__ __launch_bounds__(256) void gemm2_kernel(
    const float* __restrict__ agg, const float* __restrict__ cnt,
    const float* __restrict__ h,
    const float* __restrict__ Wl, const float* __restrict__ bl,
    const float* __restrict__ Wr, float* __restrict__ out)
{
    int wave = (blockIdx.x * 256 + threadIdx.x) >> 5;
    if (wave >= N_NODES / 16) return;
    int lane = threadIdx.x & 31;
    int half = lane >> 4;
    int l16  = lane & 15;

    int row = wave * 16 + l16;
    float scale = 1.0f / fmaxf(cnt[row], 1.0f);
    const float2* arow = (const float2*)(agg + (long)row * 128);
    const float2* hrow = (const float2*)(h   + (long)row * 128);

    v8f c[3];
#pragma unroll
    for (int nt = 0; nt < 3; ++nt) {
        int col = nt * 16 + l16;
        float bv = (col < OUT_CH) ? bl[col] : 0.0f;
#pragma unroll
        for (int r = 0; r < 8; ++r) c[nt][r] = bv;
    }

    for (int k0 = 0; k0 < 128; k0 += 4) {
        int ka = k0 + 2 * half;
        float2 av = arow[ka >> 1];
        v2f a; a.x = av.x * scale; a.y = av.y * scale;
#pragma unroll
        for (int nt = 0; nt < 3; ++nt) {
            int col = nt * 16 + l16;
            v2f b;
            b.x = (col < OUT_CH) ? Wl[ka * OUT_CH + col] : 0.0f;
            b.y = (col < OUT_CH) ? Wl[(ka + 1) * OUT_CH + col] : 0.0f;
            c[nt] = __builtin_amdgcn_wmma_f32_16x16x4_f32(
                false, a, false, b, (short)0, c[nt], false, false);
        }
    }
    for (int k0 = 0; k0 < 128; k0 += 4) {
        int ka = k0 + 2 * half;
        float2 av = hrow[ka >> 1];
        v2f a; a.x = av.x; a.y = av.y;
#pragma unroll
        for (int nt = 0; nt < 3; ++nt) {
            int col = nt * 16 + l16;
            v2f b;
            b.x = (col < OUT_CH) ? Wr[ka * OUT_CH + col] : 0.0f;
            b.y = (col < OUT_CH) ? Wr[(ka + 1) * OUT_CH + col] : 0.0f;
            c[nt] = __builtin_amdgcn_wmma_f32_16x16x4_f32(
                false, a, false, b, (short)0, c[nt], false, false);
        }
    }

    int rbase = wave * 16 + 8 * half;
#pragma unroll
    for (int nt = 0; nt < 3; ++nt) {
        int col = nt * 16 + l16;
        if (col < OUT_CH) {
#pragma unroll
            for (int r = 0; r < 8; ++r)
                out[(long)(rbase + r) * OUT_CH + col] = c[nt][r];
        }
    }
}

// ---------------- row-wise log_softmax over 40 classes, in place ------------
__global__ __launch_bounds__(256) void logsoftmax_kernel(float* __restrict__ out) {
    int row = blockIdx.x * 256 + threadIdx.x;
    if (row >= N_NODES) return;
    float* p = out + (long)row * OUT_CH;
    float v[OUT_CH];
    float m = -1e30f;
#pragma unroll
    for (int i = 0; i < OUT_CH; ++i) { v[i] = p[i]; m = fmaxf(m, v[i]); }
    float s = 0.0f;
#pragma unroll
    for (int i = 0; i < OUT_CH; ++i) s += expf(v[i] - m);
    float ls = logf(s);
#pragma unroll
    for (int i = 0; i < OUT_CH; ++i) p[i] = v[i] - m - ls;
}

// ---------------------------------------------------------------------------
extern "C" void kernel_launch(void* const* d_in, const int* in_sizes, int n_in,
                              void* d_out, int out_size, void* d_ws, size_t ws_size,
                              hipStream_t stream) {
    const float* x    = (const float*)d_in[0];
    const int*   eidx = (const int*)  d_in[1];   // [2, E]: row0=src, row1=dst
    const float* W_l1 = (const float*)d_in[2];
    const float* b_l1 = (const float*)d_in[3];
    const float* W_r1 = (const float*)d_in[4];
    const float* W_l2 = (const float*)d_in[5];
    const float* b_l2 = (const float*)d_in[6];
    const float* W_r2 = (const float*)d_in[7];
    float* out = (float*)d_out;

    const int* srcIdx = eidx;
    const int* dstIdx = eidx + N_EDGES;

    // Workspace: agg (N*128) | cnt (N) | h (N*128)   = N*257 floats (~51 MB)
    float* ws   = (float*)d_ws;
    float* agg  = ws;
    float* cnt  = ws + (long)N_NODES * 128;
    float* hbuf = cnt + N_NODES;

    const int scatterBlocks = (N_EDGES * 32) / 256;          // 100000
    const int gemmBlocks    = (N_NODES / 16 * 32 + 255) / 256; // 391

    // 1) zero agg + cnt (contiguous: N*129 floats)
    {
        int n4 = (N_NODES * 129) / 4;
        zero_kernel<<<(n4 + 255) / 256, 256, 0, stream>>>((float4*)agg, n4);
    }
    // 2) layer-1 aggregation (+ degree count)
    scatter_kernel<<<scatterBlocks, 256, 0, stream>>>(srcIdx, dstIdx, x, agg, cnt, N_EDGES);
    // 3) layer-1 fused GEMM + bias + ReLU (WMMA f32)
    gemm1_kernel<<<gemmBlocks, 256, 0, stream>>>(agg, cnt, x, W_l1, b_l1, W_r1, hbuf);
    // 4) re-zero agg for layer 2
    {
        int n4 = (N_NODES * 128) / 4;
        zero_kernel<<<(n4 + 255) / 256, 256, 0, stream>>>((float4*)agg, n4);
    }
    // 5) layer-2 aggregation of hidden features
    scatter_kernel<<<scatterBlocks, 256, 0, stream>>>(srcIdx, dstIdx, hbuf, agg, nullptr, N_EDGES);
    // 6) layer-2 fused GEMM + bias (WMMA f32) -> logits
    gemm2_kernel<<<gemmBlocks, 256, 0, stream>>>(agg, cnt, hbuf, W_l2, b_l2, W_r2, out);
    // 7) log_softmax in place
    logsoftmax_kernel<<<(N_NODES + 255) / 256, 256, 0, stream>>>(out);
}